// DeformConv2d_84121229459912
// MI455X (gfx1250) — compile-verified
//
#include <hip/hip_runtime.h>
#include <hip/hip_bf16.h>
#include <math.h>

// ---------------------------------------------------------------------------
// DeformConv2d on MI455X (gfx1250, wave32, WMMA).
//   b=2, c=64, H=W=128, KS=3, N=9, pad=1, stride=1, Hp=Wp=130
// Pipeline:
//   K0: transpose conv_w (64,576) -> Bws (576,64)            [workspace]
//   K1: 3x3 conv -> 18 offsets + 9 sigmoid modulations/pixel  [workspace]
//   K2: per 32-pixel tile: p, loss partial, bilinear-sampled A-tile in LDS,
//       GEMM (32x576)x(576x64) via v_wmma_f32_16x16x4_f32.
//       4 waves x 16 output channels; each wave runs TWO 16x16 accumulators
//       per B fragment (2 WMMA per B fetch -> halved B traffic per FLOP).
//   K3: fixed-order loss reduction -> d_out[2*64*128*128]
// ---------------------------------------------------------------------------

typedef __attribute__((ext_vector_type(2))) float v2f;
typedef __attribute__((ext_vector_type(8))) float v8f;

#define CIN   64
#define HW    128
#define NPTS  9
#define OCH   27          // 18 offset + 9 modulation channels
#define KTOT  576         // 64 * 9
#define TILE  32          // pixels per workgroup in GEMM kernel (2 M-tiles)
#define ASTR  580         // A-tile LDS row stride (mod 64 == 4 -> conflict free)
#define NPIX  32768       // 2 * 128 * 128
#define NBLK2 (NPIX / TILE)             // 1024
#define LOSS_SCALE (1.0f / 294912.0f)   // 1/(2*128*128*9)

// -------------------------------- K0 ---------------------------------------
__global__ void prep_b_kernel(const float* __restrict__ conv_w,
                              float* __restrict__ Bws) {
  int idx = blockIdx.x * blockDim.x + threadIdx.x;
  if (idx < KTOT * 64) {
    int k = idx >> 6, co = idx & 63;
    Bws[idx] = conv_w[co * KTOT + k];   // Bws[k][co]
  }
}

// -------------------------------- K1 ---------------------------------------
// One thread = one pixel; 27 f32 accumulators; weights broadcast from LDS.
__global__ __launch_bounds__(256)
void off_mod_kernel(const float* __restrict__ x,
                    const float* __restrict__ pw, const float* __restrict__ pb,
                    const float* __restrict__ mw, const float* __restrict__ mb,
                    float* __restrict__ offm) {
  __shared__ float wl[OCH * KTOT];          // 62208 B
  int t = threadIdx.x;
  for (int idx = t; idx < OCH * KTOT; idx += 256) {
    int oc = idx / KTOT, k = idx - oc * KTOT;
    wl[idx] = (oc < 18) ? pw[oc * KTOT + k] : mw[(oc - 18) * KTOT + k];
  }
  __syncthreads();

  int gpix = blockIdx.x * 256 + t;          // 0..32767
  int b  = gpix >> 14;
  int pl = gpix & 16383;
  int i  = pl >> 7, j = pl & 127;

  float acc[OCH];
#pragma unroll
  for (int o = 0; o < OCH; ++o) acc[o] = 0.f;

  const float* xb = x + (size_t)b * CIN * HW * HW;
  for (int ci = 0; ci < CIN; ++ci) {
    const float* xc = xb + (size_t)ci * HW * HW;
#pragma unroll
    for (int dy = 0; dy < 3; ++dy) {
      int yy = i + dy - 1;
#pragma unroll
      for (int dx = 0; dx < 3; ++dx) {
        int xx = j + dx - 1;
        float xv = (yy >= 0 && yy < HW && xx >= 0 && xx < HW)
                       ? xc[yy * HW + xx] : 0.f;
        int k = ci * 9 + dy * 3 + dx;
#pragma unroll
        for (int o = 0; o < OCH; ++o)
          acc[o] = fmaf(wl[o * KTOT + k], xv, acc[o]);
      }
    }
  }

  float* dst = offm + (size_t)gpix * OCH;
#pragma unroll
  for (int o = 0; o < 18; ++o) dst[o] = acc[o] + pb[o];
#pragma unroll
  for (int o = 0; o < 9; ++o) {
    float v = acc[18 + o] + mb[o];
    dst[18 + o] = 1.f / (1.f + __expf(-v));   // sigmoid
  }
}

// -------------------------------- K2 ---------------------------------------
__device__ __forceinline__ float sampleXp(const float* __restrict__ xc,
                                          int qx, int qy) {
  // padded coords [0,129] -> x at (qx-1, qy-1); zero padding
  qx -= 1; qy -= 1;
  if ((unsigned)qx < 128u && (unsigned)qy < 128u) return xc[qx * HW + qy];
  return 0.f;
}

__global__ __launch_bounds__(128)
void deform_gemm_kernel(const float* __restrict__ x,
                        const float* __restrict__ offm,
                        const float* __restrict__ Bws,
                        const float* __restrict__ centers,
                        float* __restrict__ out,
                        float* __restrict__ partials) {
  __shared__ float A[TILE * ASTR];            // 74240 B im2col tile (32 x 576)
  __shared__ int   meta_i[TILE * NPTS * 4];   // qlt_x, qlt_y, qrb_x, qrb_y
  __shared__ float meta_f[TILE * NPTS * 4];   // g_lt*m, g_rb*m, g_lb*m, g_rt*m
  __shared__ float lsum[128];

  int t = threadIdx.x;
  int gbase  = blockIdx.x * TILE;             // first global pixel of tile
  int b      = gbase >> 14;
  int plbase = gbase & 16383;                 // pixel within image

  // ---- Stage A: sampling coordinates, bilinear weights, loss --------------
  float myloss = 0.f;
  for (int pp = t; pp < TILE * NPTS; pp += 128) {
    int pix = pp / 9, n = pp - pix * 9;
    int gpix = gbase + pix;
    int pl = plbase + pix;
    int i = pl >> 7, j = pl & 127;
    const float* om = offm + (size_t)gpix * OCH;
    float offx = om[n], offy = om[9 + n], mv = om[18 + n];
    float px = (float)(i + 1) + (float)(n / 3 - 1) + offx;
    float py = (float)(j + 1) + (float)(n % 3 - 1) + offy;

    // loss: min squared distance to 4 centers (unclipped p)
    float dmin = 3.4e38f;
#pragma unroll
    for (int kk = 0; kk < 4; ++kk) {
      float dx = px - centers[kk * 2 + 0];
      float dy = py - centers[kk * 2 + 1];
      dmin = fminf(dmin, dx * dx + dy * dy);
    }
    myloss += dmin;

    float fx = floorf(px), fy = floorf(py);
    const float mx = 129.f;                   // Hp-1 = Wp-1
    float qltx = fminf(fmaxf(fx,       0.f), mx);
    float qlty = fminf(fmaxf(fy,       0.f), mx);
    float qrbx = fminf(fmaxf(fx + 1.f, 0.f), mx);
    float qrby = fminf(fmaxf(fy + 1.f, 0.f), mx);
    float pxc  = fminf(fmaxf(px, 0.f), mx);
    float pyc  = fminf(fmaxf(py, 0.f), mx);
    float glt = (1.f + (qltx - pxc)) * (1.f + (qlty - pyc));
    float grb = (1.f - (qrbx - pxc)) * (1.f - (qrby - pyc));
    float glb = (1.f + (qltx - pxc)) * (1.f - (qrby - pyc));
    float grt = (1.f - (qrbx - pxc)) * (1.f + (qlty - pyc));

    int mi = pp * 4;
    meta_i[mi + 0] = (int)qltx; meta_i[mi + 1] = (int)qlty;
    meta_i[mi + 2] = (int)qrbx; meta_i[mi + 3] = (int)qrby;
    meta_f[mi + 0] = glt * mv;  meta_f[mi + 1] = grb * mv;
    meta_f[mi + 2] = glb * mv;  meta_f[mi + 3] = grt * mv;
  }
  lsum[t] = myloss;
  __syncthreads();
  if (t == 0) {                               // fixed order -> deterministic
    float s = 0.f;
    for (int k = 0; k < 128; ++k) s += lsum[k];
    partials[blockIdx.x] = s;
  }

  // ---- Stage B: bilinear gather -> A tile (32 x 576) in LDS ---------------
  for (int col = t; col < KTOT; col += 128) {
    int ci = col / 9, n = col - ci * 9;
    const float* xc = x + (size_t)(b * CIN + ci) * (HW * HW);
    for (int pix = 0; pix < TILE; ++pix) {
      int mi = (pix * 9 + n) * 4;
      int ax = meta_i[mi], ay = meta_i[mi + 1];
      int bx = meta_i[mi + 2], by = meta_i[mi + 3];
      float xo = meta_f[mi + 0] * sampleXp(xc, ax, ay)
               + meta_f[mi + 1] * sampleXp(xc, bx, by)
               + meta_f[mi + 2] * sampleXp(xc, ax, by)
               + meta_f[mi + 3] * sampleXp(xc, bx, ay);
      A[pix * ASTR + col] = xo;
    }
  }
  __syncthreads();

  // ---- Stage C: WMMA GEMM, wave w owns output channels [16w, 16w+16);
  //      two 16-row M-tiles share each B fragment (2 WMMA per B fetch). -----
  int wave = t >> 5, lane = t & 31;
  int co0  = wave * 16;
  int rowA = lane & 15;                       // M row within tile
  int koff = (lane >> 4) * 2;                 // K sub-offset per lane half
  v8f acc0 = {};                              // pixels plbase +  0..15
  v8f acc1 = {};                              // pixels plbase + 16..31
  for (int k = 0; k < KTOT; k += 4) {
    v2f a0, a1, bf;
    a0.x = A[rowA * ASTR + k + koff];         // A frag: 16x4 f32 (ISA 7.12.2)
    a0.y = A[rowA * ASTR + k + koff + 1];
    a1.x = A[(rowA + 16) * ASTR + k + koff];
    a1.y = A[(rowA + 16) * ASTR + k + koff + 1];
    bf.x = Bws[(k + koff)     * 64 + co0 + rowA];   // B frag: 4x16 f32
    bf.y = Bws[(k + koff + 1) * 64 + co0 + rowA];
    acc0 = __builtin_amdgcn_wmma_f32_16x16x4_f32(
        false, a0, false, bf, (short)0, acc0, false, false);
    acc1 = __builtin_amdgcn_wmma_f32_16x16x4_f32(
        false, a1, false, bf, (short)0, acc1, false, false);
  }

  // D layout: VGPR v -> M = v (lanes 0-15) / v+8 (lanes 16-31); N = lane&15
  int co    = co0 + rowA;
  int mbase = (lane >> 4) * 8;
  size_t outBase = (size_t)(b * CIN + co) * (HW * HW) + plbase;
#pragma unroll
  for (int v = 0; v < 8; ++v)
    out[outBase + mbase + v] = acc0[v];
#pragma unroll
  for (int v = 0; v < 8; ++v)
    out[outBase + 16 + mbase + v] = acc1[v];
}

// -------------------------------- K3 ---------------------------------------
__global__ __launch_bounds__(256)
void loss_reduce_kernel(const float* __restrict__ partials,
                        float* __restrict__ loss_out) {
  __shared__ float s[256];
  int t = threadIdx.x;
  float v = 0.f;
  for (int idx = t; idx < NBLK2; idx += 256) v += partials[idx];
  s[t] = v;
  __syncthreads();
  if (t == 0) {
    float tot = 0.f;
    for (int k = 0; k < 256; ++k) tot += s[k];
    *loss_out = tot * LOSS_SCALE;
  }
}

// ---------------------------------------------------------------------------
extern "C" void kernel_launch(void* const* d_in, const int* in_sizes, int n_in,
                              void* d_out, int out_size, void* d_ws, size_t ws_size,
                              hipStream_t stream) {
  const float* x        = (const float*)d_in[0];   // (2,64,128,128)
  const float* centers  = (const float*)d_in[1];   // (4,2)
  const float* p_conv_w = (const float*)d_in[2];   // (18,64,3,3)
  const float* p_conv_b = (const float*)d_in[3];   // (18,)
  const float* m_conv_w = (const float*)d_in[4];   // (9,64,3,3)
  const float* m_conv_b = (const float*)d_in[5];   // (9,)
  const float* conv_w   = (const float*)d_in[6];   // (64,64,3,3)

  float* out = (float*)d_out;                      // 2*64*128*128 + 1 floats
  float* loss_slot = out + (size_t)2 * 64 * 128 * 128;

  // workspace layout (floats): offm[32768*27] | Bws[576*64] | partials[1024]
  float* ws       = (float*)d_ws;
  float* offm     = ws;
  float* Bws      = offm + (size_t)NPIX * OCH;
  float* partials = Bws + (size_t)KTOT * 64;

  prep_b_kernel<<<(KTOT * 64 + 255) / 256, 256, 0, stream>>>(conv_w, Bws);
  off_mod_kernel<<<NPIX / 256, 256, 0, stream>>>(x, p_conv_w, p_conv_b,
                                                 m_conv_w, m_conv_b, offm);
  deform_gemm_kernel<<<NBLK2, 128, 0, stream>>>(x, offm, Bws, centers,
                                                out, partials);
  loss_reduce_kernel<<<1, 256, 0, stream>>>(partials, loss_slot);
}